// SlidingWindowAttention_47218870453071
// MI455X (gfx1250) — compile-verified
//
#include <hip/hip_runtime.h>
#include <hip/hip_bf16.h>

// ---------------------------------------------------------------------------
// Sliding-window attention, fp32, CDNA5 (gfx1250, wave32).
// Matrix ops use V_WMMA_F32_16X16X4_F32 (native fp32 WMMA).
//
// WMMA f32 16x16x4 register layouts (CDNA5 ISA 7.12.2):
//   A (16x4, MxK):  lane l: m = l&15; VGPR0 = A[m][2*(l>>4)+0], VGPR1 = A[m][2*(l>>4)+1]
//   B (4x16, KxN):  lane l: n = l&15; VGPR0 = B[2*(l>>4)+0][n], VGPR1 = B[2*(l>>4)+1][n]
//   C/D (16x16):    lane l: n = l&15; VGPRv = C[v + 8*(l>>4)][n]
// ---------------------------------------------------------------------------

typedef __attribute__((ext_vector_type(2))) float v2f;
typedef __attribute__((ext_vector_type(4))) float v4f;
typedef __attribute__((ext_vector_type(8))) float v8f;

#define NUMH    16
#define HDIM    64
#define WINDOW  512
#define DM      1024
#define BATCH   2
#define SEQ     2048
#define MTOT    (BATCH * SEQ)      // 4096 rows total

#define BAND_W  544                // 34 key tiles of 16 cover [i0-512, i0+31]
#define BAND_S  548                // padded LDS stride (548*8 % 64 != 0 -> no bank clash)
#define QPAD    68                 // padded Q-tile stride
#define KSTR    68                 // K staging stride: banks 4*lm+d+c hit all 64 banks once
#define VSTR    72                 // V staging stride: +2-row half-wave offset = +16 banks

// ===========================================================================
// Generic LDS-tiled GEMM: Y[4096][1024] = X[4096][1024] @ W[1024][1024] + bias
// Block tile 128x128, BK=32, 8 waves, each wave owns a 32x64 sub-tile
// (2x4 WMMA accumulators). 64 wmma per k-chunk. Next-chunk lines are
// prefetched (global_prefetch_b8) so HBM/L2 fetch overlaps the WMMA burst.
// ===========================================================================
__launch_bounds__(256)
__global__ void gemm_bias_kernel(const float* __restrict__ X,
                                 const float* __restrict__ W,
                                 const float* __restrict__ bias,
                                 float* __restrict__ Y)
{
    const int K = DM, N = DM;
    __shared__ float As[128][36];   // As[m][k], padded stride 36
    __shared__ float Bs[128][36];   // Bs[n][k] (W tile stored transposed)

    const int t    = threadIdx.x;
    const int lane = t & 31;
    const int wave = t >> 5;
    const int lm   = lane & 15;
    const int hi   = lane >> 4;
    const int m0   = blockIdx.y * 128;
    const int n0   = blockIdx.x * 128;
    const int wrow = wave >> 1;      // 0..3 -> rows wrow*32
    const int wcol = wave & 1;       // 0..1 -> cols wcol*64

    v8f acc[2][4];
    const v8f vzero8 = {0.f,0.f,0.f,0.f,0.f,0.f,0.f,0.f};
#pragma unroll
    for (int i = 0; i < 2; ++i)
#pragma unroll
        for (int j = 0; j < 4; ++j) acc[i][j] = vzero8;

    for (int kb = 0; kb < K; kb += 32) {
        // Stage A tile (128x32): 1024 float4, 4 per thread, coalesced
#pragma unroll
        for (int f = 0; f < 4; ++f) {
            int flat = t + f * 256;
            int r  = flat >> 3;        // 8 float4 per row
            int c4 = flat & 7;
            v4f a = *(const v4f*)(X + (size_t)(m0 + r) * K + kb + c4 * 4);
            *(v4f*)(&As[r][c4 * 4]) = a;
        }
        // Stage B tile (32x128) transposed into Bs[n][k]
#pragma unroll
        for (int f = 0; f < 4; ++f) {
            int flat = t + f * 256;
            int kr = flat >> 5;        // 32 float4 per row
            int c4 = flat & 31;
            v4f w4 = *(const v4f*)(W + (size_t)(kb + kr) * N + n0 + c4 * 4);
#pragma unroll
            for (int jj = 0; jj < 4; ++jj) Bs[c4 * 4 + jj][kr] = w4[jj];
        }
        __syncthreads();

        // Prefetch next k-chunk while this chunk's 64 WMMAs execute
        if (kb + 32 < K) {
            __builtin_prefetch(X + (size_t)(m0 + (t >> 3)) * K + (kb + 32) + (t & 7) * 4, 0, 3);
            __builtin_prefetch(W + (size_t)(kb + 32 + (t >> 5)) * N + n0 + lane * 4, 0, 3);
        }

#pragma unroll
        for (int kk = 0; kk < 32; kk += 4) {
            v2f afrag[2], bfrag[4];
#pragma unroll
            for (int mt = 0; mt < 2; ++mt)
                afrag[mt] = *(const v2f*)(&As[wrow * 32 + mt * 16 + lm][kk + 2 * hi]);
#pragma unroll
            for (int nt = 0; nt < 4; ++nt)
                bfrag[nt] = *(const v2f*)(&Bs[wcol * 64 + nt * 16 + lm][kk + 2 * hi]);
#pragma unroll
            for (int mt = 0; mt < 2; ++mt)
#pragma unroll
                for (int nt = 0; nt < 4; ++nt)
                    acc[mt][nt] = __builtin_amdgcn_wmma_f32_16x16x4_f32(
                        false, afrag[mt], false, bfrag[nt],
                        (short)0, acc[mt][nt], false, false);
        }
        __syncthreads();
    }

    // Epilogue: bias + store
#pragma unroll
    for (int nt = 0; nt < 4; ++nt) {
        int col = n0 + wcol * 64 + nt * 16 + lm;
        float bv = bias[col];
#pragma unroll
        for (int mt = 0; mt < 2; ++mt) {
#pragma unroll
            for (int v = 0; v < 8; ++v) {
                int row = m0 + wrow * 32 + mt * 16 + v + 8 * hi;
                Y[(size_t)row * N + col] = acc[mt][nt][v] + bv;
            }
        }
    }
}

// ===========================================================================
// Banded sliding-window attention. One block = (b, h, 32 query rows).
// Band = 34 key tiles (544 cols) covering j in [i0-512, i0+31].
//  P2: scores 32x544 via WMMA; K staged 64 rows at a time into LDS (coalesced)
//  P3: row softmax over band (8 threads/row), normalize in LDS
//  P3b: stream full 2048-wide probability rows (zeros outside band), NT stores
//  P4: attended (32x64) = P @ V via WMMA; V staged 64 rows at a time into LDS
// K and V staging share one LDS buffer (phases are barrier-separated).
// ===========================================================================
__launch_bounds__(256)
__global__ void swa_attention_kernel(const float* __restrict__ Qm,
                                     const float* __restrict__ Km,
                                     const float* __restrict__ Vm,
                                     float* __restrict__ attw,
                                     float* __restrict__ attn_out)
{
    __shared__ float Qs[32][QPAD];
    __shared__ float Pband[32][BAND_S];
    __shared__ float KVs[64 * VSTR];    // K (stride KSTR) in P2, V (stride VSTR) in P4
    __shared__ float redA[32][8];
    __shared__ float redB[32][8];

    const int t    = threadIdx.x;
    const int lane = t & 31;
    const int wave = t >> 5;
    const int lm   = lane & 15;
    const int hi   = lane >> 4;
    const int wy   = wave >> 2;          // 0..1 row half
    const int wx   = wave & 3;           // 0..3 tile column group

    const int i0  = blockIdx.x * 32;                  // query tile start
    const int bh  = blockIdx.y;                       // b*16 + h
    const int b   = bh >> 4;
    const int h   = bh & 15;
    const int jb0 = i0 - 512;                         // band global col start
    const size_t rowbase = (size_t)(b * SEQ) * DM + (size_t)h * HDIM;

    const v8f vzero8 = {0.f,0.f,0.f,0.f,0.f,0.f,0.f,0.f};
    const v4f vzero4 = {0.f,0.f,0.f,0.f};

    // ---- P1: load Q tile (32x64), pre-scaled by 1/sqrt(64) ----
#pragma unroll
    for (int f = 0; f < 2; ++f) {
        int flat = t + f * 256;          // 512 float4
        int r  = flat >> 4;              // 16 float4 per row
        int c4 = flat & 15;
        v4f q4 = *(const v4f*)(Qm + rowbase + (size_t)(i0 + r) * DM + c4 * 4);
        q4 *= 0.125f;
        *(v4f*)(&Qs[r][c4 * 4]) = q4;
    }
    __syncthreads();

    // ---- P2: scores into band; stage 4 key tiles (64 rows) per group ----
    for (int g = 0; g < 9; ++g) {
        int jbase = jb0 + g * 64;
        // Coalesced staging of K rows [jbase, jbase+64) into KVs (stride KSTR)
#pragma unroll
        for (int f = 0; f < 4; ++f) {
            int flat = t + f * 256;      // 1024 float4
            int r  = flat >> 4;          // 16 float4 per row
            int c4 = flat & 15;
            int jr = jbase + r;
            jr = jr < 0 ? 0 : (jr > SEQ - 1 ? SEQ - 1 : jr);
            v4f kv = *(const v4f*)(Km + rowbase + (size_t)jr * DM + c4 * 4);
            *(v4f*)(&KVs[r * KSTR + c4 * 4]) = kv;
        }
        __syncthreads();

        int tile = g * 4 + wx;           // wave-uniform guard keeps EXEC full
        if (tile < 34) {
            int j0 = jb0 + tile * 16;
            v8f c = vzero8;
#pragma unroll
            for (int d = 0; d < HDIM; d += 4) {
                v2f a  = *(const v2f*)(&Qs[wy * 16 + lm][d + 2 * hi]);
                v2f bf = *(const v2f*)(&KVs[(wx * 16 + lm) * KSTR + d + 2 * hi]);
                c = __builtin_amdgcn_wmma_f32_16x16x4_f32(
                        false, a, false, bf, (short)0, c, false, false);
            }
#pragma unroll
            for (int v = 0; v < 8; ++v) {
                int i    = i0 + wy * 16 + v + 8 * hi;
                int j    = j0 + lm;
                int diff = i - j;
                bool keep = (j >= 0) && (diff >= 0) && (diff < WINDOW);
                Pband[wy * 16 + v + 8 * hi][tile * 16 + lm] = keep ? c[v] : -1.0e9f;
            }
        }
        __syncthreads();
    }

    // ---- P3: softmax over 544-wide band; 8 threads per row, 68 cols each ----
    const int row  = t >> 3;   // 0..31
    const int part = t & 7;
    const int c0   = part * 68;

    float mx = -3.0e38f;
    for (int c = 0; c < 68; ++c) mx = fmaxf(mx, Pband[row][c0 + c]);
    redA[row][part] = mx;
    __syncthreads();
    float rmax = redA[row][0];
#pragma unroll
    for (int p = 1; p < 8; ++p) rmax = fmaxf(rmax, redA[row][p]);

    float s = 0.f;
    for (int c = 0; c < 68; ++c) {
        float e = __expf(Pband[row][c0 + c] - rmax);   // masked -> exp(-1e9) == 0
        Pband[row][c0 + c] = e;
        s += e;
    }
    redB[row][part] = s;
    __syncthreads();
    float tot = 0.f;
#pragma unroll
    for (int p = 0; p < 8; ++p) tot += redB[row][p];
    float inv = 1.f / tot;
    for (int c = 0; c < 68; ++c) Pband[row][c0 + c] *= inv;
    __syncthreads();

    // ---- P3b: stream full 2048-wide weight rows (537 MB total = HBM floor).
    // 8 threads/row write interleaved 16B chunks -> 128B contiguous per row per
    // iteration; non-temporal so the stream doesn't evict Q/K/V from L2.
    {
        float* wp = attw + ((size_t)bh * SEQ + (size_t)(i0 + row)) * SEQ;
#pragma unroll 4
        for (int f = 0; f < 64; ++f) {
            int j = part * 4 + f * 32;
            int c = j - jb0;
            v4f val = (c >= 0 && c < BAND_W) ? *(const v4f*)(&Pband[row][c]) : vzero4;
            __builtin_nontemporal_store(val, (v4f*)(wp + j));
        }
    }

    // ---- P4: attended = P @ V ; V staged 64 rows per chunk (last chunk 32).
    // Per-lane B reads hit rows k,k+1 (VSTR dwords apart) -> ds_load_2addr_b32.
    v8f acc = vzero8;
    for (int ch = 0; ch < 9; ++ch) {
        const int nk = (ch == 8) ? 32 : 64;          // 544 = 8*64 + 32
        int jbase = jb0 + ch * 64;
        for (int f = 0; f < (nk >> 4); ++f) {        // wave-uniform trip count
            int flat = t + f * 256;
            int r  = flat >> 4;
            int c4 = flat & 15;
            int jr = jbase + r;
            jr = jr < 0 ? 0 : (jr > SEQ - 1 ? SEQ - 1 : jr);
            v4f vv = *(const v4f*)(Vm + rowbase + (size_t)jr * DM + c4 * 4);
            *(v4f*)(&KVs[r * VSTR + c4 * 4]) = vv;
        }
        __syncthreads();

        for (int kkl = 0; kkl < nk; kkl += 4) {
            v2f a = *(const v2f*)(&Pband[wy * 16 + lm][ch * 64 + kkl + 2 * hi]);
            int r0 = kkl + 2 * hi;
            v2f bf;
            bf[0] = KVs[r0 * VSTR + wx * 16 + lm];
            bf[1] = KVs[(r0 + 1) * VSTR + wx * 16 + lm];
            acc = __builtin_amdgcn_wmma_f32_16x16x4_f32(
                      false, a, false, bf, (short)0, acc, false, false);
        }
        __syncthreads();
    }

#pragma unroll
    for (int v = 0; v < 8; ++v) {
        int r = i0 + wy * 16 + v + 8 * hi;
        attn_out[rowbase + (size_t)r * DM + wx * 16 + lm] = acc[v];
    }
}

// ===========================================================================
extern "C" void kernel_launch(void* const* d_in, const int* in_sizes, int n_in,
                              void* d_out, int out_size, void* d_ws, size_t ws_size,
                              hipStream_t stream) {
    (void)in_sizes; (void)n_in; (void)out_size; (void)ws_size;

    const float* x  = (const float*)d_in[0];
    const float* Wq = (const float*)d_in[1];
    const float* bq = (const float*)d_in[2];
    const float* Wk = (const float*)d_in[3];
    const float* bk = (const float*)d_in[4];
    const float* Wv = (const float*)d_in[5];
    const float* bv = (const float*)d_in[6];
    const float* Wo = (const float*)d_in[7];
    const float* bo = (const float*)d_in[8];

    float* out  = (float*)d_out;                                   // (2,2048,1024)
    float* attw = out + (size_t)BATCH * SEQ * DM;                  // (2,16,2048,2048)

    const size_t mat = (size_t)MTOT * DM;                          // 16 MB each
    float* q  = (float*)d_ws;
    float* k  = q  + mat;
    float* v  = k  + mat;
    float* ao = v  + mat;                                          // 64 MB workspace

    dim3 gthreads(256);
    dim3 ggrid(DM / 128, MTOT / 128);                              // (8, 32)

    gemm_bias_kernel<<<ggrid, gthreads, 0, stream>>>(x, Wq, bq, q);
    gemm_bias_kernel<<<ggrid, gthreads, 0, stream>>>(x, Wk, bk, k);
    gemm_bias_kernel<<<ggrid, gthreads, 0, stream>>>(x, Wv, bv, v);

    swa_attention_kernel<<<dim3(SEQ / 32, BATCH * NUMH), gthreads, 0, stream>>>(
        q, k, v, attw, ao);

    gemm_bias_kernel<<<ggrid, gthreads, 0, stream>>>(ao, Wo, bo, out);
}